// TSEncoder_m_alt_62311385531012
// MI455X (gfx1250) — compile-verified
//
#include <hip/hip_runtime.h>
#include <hip/hip_bf16.h>
#include <math.h>

// ---------------------------------------------------------------------------
// Problem constants (from the reference)
// ---------------------------------------------------------------------------
#define BB      16
#define TT      512
#define NMED    13
#define INDIMS  130
#define HIDDEN  256
#define OUTPUT  320
#define DEPTH   10

typedef __attribute__((ext_vector_type(2))) float v2f;
typedef __attribute__((ext_vector_type(8))) float v8f;

__device__ __forceinline__ float gelu_exact(float x) {
    return 0.5f * x * (1.0f + erff(x * 0.70710678118654752440f));
}

// ---------------------------------------------------------------------------
// Kernel 1: sparse scatter (13 meds) + NaN mask + FC 130->256, store (B,256,T)
// One block per (t, b); 256 threads = HIDDEN. Serial scatter preserves the
// reference's "set" (overwrite) semantics on duplicate ids.
// ---------------------------------------------------------------------------
__global__ __launch_bounds__(256) void scatter_fc_kernel(
    const float* __restrict__ x0, const float* __restrict__ emb,
    const float* __restrict__ fcw, const float* __restrict__ fcb,
    float* __restrict__ out)
{
    const int t = blockIdx.x;
    const int b = blockIdx.y;
    const int h = threadIdx.x;

    __shared__ float meds[INDIMS + 1];
    __shared__ int   sids[NMED];
    __shared__ int   scnt;
    __shared__ int   sflag;
    __shared__ unsigned char wfl[INDIMS + 1];

    if (h < INDIMS + 1) { meds[h] = 0.0f; wfl[h] = 0; }
    if (h == 0) { scnt = 0; sflag = 0; }
    __syncthreads();

    if (h == 0) {
        const float* row = x0 + ((size_t)b * TT + t) * (3 * NMED);
        int cnt = 0;
        for (int m = 0; m < NMED; ++m) {
            float dose = row[NMED + m];
            int unit   = (int)row[2 * NMED + m];
            float val  = dose * emb[unit];
            int id     = (dose != 0.0f) ? (int)row[m] : INDIMS; // 130 = dropped col
            meds[id] = val;                                     // overwrite: last wins
            if (id < INDIMS && !wfl[id]) { wfl[id] = 1; sids[cnt++] = id; }
        }
        scnt = cnt;
    }
    __syncthreads();
    if (h < INDIMS) { float v = meds[h]; if (!(v == v)) sflag = 1; }
    __syncthreads();

    float acc = fcb[h];
    if (!sflag) {
        int n = scnt;
        for (int j = 0; j < n; ++j) {
            int id = sids[j];
            acc = fmaf(meds[id], fcw[(size_t)id * HIDDEN + h], acc);
        }
    }
    out[((size_t)b * HIDDEN + h) * TT + t] = acc;
}

// ---------------------------------------------------------------------------
// Kernel 2: repack weights (Cout,Cin,K) -> pair-interleaved (K, Cin/2, Cout, 2)
// so each lane's (ci, ci+1) A-operand pair is one contiguous b64 load,
// coalesced over Cout across lanes. (Cin is even for all layers.)
// ---------------------------------------------------------------------------
__global__ __launch_bounds__(256) void repack_w_kernel(
    const float* __restrict__ in, float* __restrict__ out,
    int Cout, int Cin, int K)
{
    int idx = blockIdx.x * 256 + threadIdx.x;
    int total = Cout * Cin * K;
    if (idx >= total) return;
    int p   = idx & 1;              // element within pair
    int r   = idx >> 1;
    int co  = r % Cout;
    int r2  = r / Cout;
    int ci2 = r2 % (Cin >> 1);
    int k   = r2 / (Cin >> 1);
    int ci  = (ci2 << 1) | p;
    out[idx] = in[((size_t)co * Cin + ci) * K + k];
}

// ---------------------------------------------------------------------------
// Kernel 3: dilated 1-D conv as fp32 WMMA GEMM (V_WMMA_F32_16X16X4_F32),
// fully specialized at compile time so the K-loop unrolls with immediate
// load offsets.
//   y[b,co,t] = bias[co] + sum_k sum_ci W[co,ci,k] * f(x[b,ci,t+(k-c)*d])
// Workgroup = (32-column time tile, batch), 8 waves.
// Tap-outer loop: stage f(x) for one tap into LDS (pair-interleaved,
// CIN*32 floats <= 40KB); every wave updates its owned M-tiles.
// Each A b64 load feeds two WMMAs (two 16-col N-subtiles).
// Accumulators for <=3 owned M-tiles persist across taps (uniform barriers).
// ---------------------------------------------------------------------------
template<int CIN, int COUT, int NTAPS, bool GELU, bool ADDRES, bool TRANS>
__global__ __launch_bounds__(256) void conv_wmma_kernel(
    const float* __restrict__ xg,     // (B, CIN, T)
    const float* __restrict__ wp2,    // pair-interleaved (NTAPS, CIN/2, COUT, 2)
    const float* __restrict__ bias,   // (COUT)
    const float* __restrict__ addsrc, // (B, COUT, T) or unused
    float* __restrict__ out,
    int dil)
{
    constexpr int CIN2   = CIN >> 1;
    constexpr int MTILES = COUT >> 4;      // 16 or 20
    constexpr int CENTER = NTAPS >> 1;
    constexpr int NELEM  = CIN << 5;       // staged elements per tap

    __shared__ float ldsB[CIN * 32];       // ldsB[(ci>>1)*64 + n*2 + (ci&1)]

    const int tb   = blockIdx.x << 5;      // 32-column time base
    const int b    = blockIdx.y;
    const int tid  = threadIdx.x;
    const int wave = tid >> 5;
    const int lane = tid & 31;
    const int laneH = lane >> 4;           // K-pair select
    const int laneL = lane & 15;           // A: row M; B/D: col N

    const v8f vzero = {0.f, 0.f, 0.f, 0.f, 0.f, 0.f, 0.f, 0.f};
    v8f acc[3][2];
    #pragma unroll
    for (int j = 0; j < 3; ++j) { acc[j][0] = vzero; acc[j][1] = vzero; }

    for (int tap = 0; tap < NTAPS; ++tap) {
        __syncthreads();   // previous tap's LDS fully consumed
        // ---- stage f(x) for this tap (zero outside [0,T)) ----
        const int toff = tb + (tap - CENTER) * dil;
        const float* __restrict__ xb = xg + (size_t)b * CIN * TT;
        #pragma unroll 4
        for (int idx = tid; idx < NELEM; idx += 256) {
            int n  = idx & 31;
            int ci = idx >> 5;
            int t  = toff + n;
            float v = 0.0f;
            if ((unsigned)t < (unsigned)TT) {
                v = xb[(size_t)ci * TT + t];
                if (GELU) v = gelu_exact(v);
            }
            ldsB[((ci >> 1) << 6) + (n << 1) + (ci & 1)] = v;
        }
        const float* __restrict__ wtap = wp2 + (size_t)tap * CIN2 * COUT * 2;
        if (tap + 1 < NTAPS)   // pull next tap's weight block toward L2
            __builtin_prefetch(wtap + (size_t)CIN2 * COUT * 2, 0, 2);
        __syncthreads();

        // ---- accumulate: per owned M-tile, K-loop in steps of 4 channels ----
        #pragma unroll
        for (int j = 0; j < 3; ++j) {
            const int mt = wave + (j << 3);
            if (8 * j >= MTILES) break;        // compile-time fold for MTILES=16
            if (mt >= MTILES) continue;        // runtime guard only when MTILES=20
            const int coA = (mt << 4) + laneL; // A row (M = lane&15)
            const float* __restrict__ wrow = wtap + ((size_t)laneH * COUT + coA) * 2;
            const float* __restrict__ lrow = &ldsB[(laneH << 6) + (laneL << 1)];
            v8f a0 = acc[j][0];
            v8f a1 = acc[j][1];
            #pragma unroll 8
            for (int s = 0; s < CIN2 / 2; ++s) {
                v2f a  = *(const v2f*)(wrow + (size_t)s * (COUT << 2));
                v2f b0 = *(const v2f*)(lrow + (s << 7));
                v2f b1 = *(const v2f*)(lrow + (s << 7) + 32);
                a0 = __builtin_amdgcn_wmma_f32_16x16x4_f32(
                        false, a, false, b0, (short)0, a0, false, false);
                a1 = __builtin_amdgcn_wmma_f32_16x16x4_f32(
                        false, a, false, b1, (short)0, a1, false, false);
            }
            acc[j][0] = a0;
            acc[j][1] = a1;
        }
    }

    // ---- epilogue: bias (+ residual), store ----
    #pragma unroll
    for (int j = 0; j < 3; ++j) {
        const int mt = wave + (j << 3);
        if (8 * j >= MTILES) break;
        if (mt >= MTILES) continue;
        const int rowb = (mt << 4) + (laneH << 3);
        const float4 bq0 = *(const float4*)(bias + rowb);      // rows rowb..rowb+3
        const float4 bq1 = *(const float4*)(bias + rowb + 4);  // rows rowb+4..rowb+7
        const float bl[8] = {bq0.x, bq0.y, bq0.z, bq0.w, bq1.x, bq1.y, bq1.z, bq1.w};
        #pragma unroll
        for (int half = 0; half < 2; ++half) {
            const int col = tb + (half << 4) + laneL;
            #pragma unroll
            for (int r = 0; r < 8; ++r) {
                const int row = rowb + r;                      // D: M = r (+8 upper half)
                float y = acc[j][half][r] + bl[r];
                if (ADDRES) y += addsrc[((size_t)b * COUT + row) * TT + col];
                if (TRANS)
                    out[((size_t)b * TT + col) * COUT + row] = y;  // (B,T,COUT)
                else
                    out[((size_t)b * COUT + row) * TT + col] = y;  // (B,COUT,T)
            }
        }
    }
}

// ---------------------------------------------------------------------------
// Host-side launch sequence
// d_in order: x0, embed_table, fc_w, fc_b,
//             conv_ws flattened (w1_0,w2_0,...,w1_10,w2_10)  -> d_in[4..25]
//             conv_bs flattened (b1_0,b2_0,...,b1_10,b2_10)  -> d_in[26..47]
//             proj_w -> d_in[48], proj_b -> d_in[49]
// ---------------------------------------------------------------------------
extern "C" void kernel_launch(void* const* d_in, const int* in_sizes, int n_in,
                              void* d_out, int out_size, void* d_ws, size_t ws_size,
                              hipStream_t stream) {
    (void)in_sizes; (void)n_in; (void)out_size; (void)ws_size;

    const float* x0    = (const float*)d_in[0];
    const float* emb   = (const float*)d_in[1];
    const float* fcw   = (const float*)d_in[2];
    const float* fcb   = (const float*)d_in[3];
    const float* projw = (const float*)d_in[48];
    const float* projb = (const float*)d_in[49];

    float* ws = (float*)d_ws;
    const size_t BUF = (size_t)BB * OUTPUT * TT;   // 2,621,440 floats (10 MB)
    float* bufX = ws;             // activation ping
    float* bufA = ws + BUF;       // inner intermediate
    float* bufB = ws + 2 * BUF;   // activation pong
    float* bufR = ws + 3 * BUF;   // projection residual
    float* wts  = ws + 4 * BUF;   // repacked weights

    // ---- repack all conv weights to pair-interleaved (K, Cin/2, Cout, 2) ----
    float* w1t[DEPTH + 1];
    float* w2t[DEPTH + 1];
    size_t o = 0;
    for (int i = 0; i <= DEPTH; ++i) {
        int cout = (i == DEPTH) ? OUTPUT : HIDDEN;
        int cin1 = HIDDEN;   // w1 input channels are always 256
        int cin2 = cout;     // w2 input channels = cout
        {
            int total = cout * cin1 * 3;
            w1t[i] = wts + o;
            repack_w_kernel<<<(total + 255) / 256, 256, 0, stream>>>(
                (const float*)d_in[4 + 2 * i], w1t[i], cout, cin1, 3);
            o += (size_t)total;
        }
        {
            int total = cout * cin2 * 3;
            w2t[i] = wts + o;
            repack_w_kernel<<<(total + 255) / 256, 256, 0, stream>>>(
                (const float*)d_in[5 + 2 * i], w2t[i], cout, cin2, 3);
            o += (size_t)total;
        }
    }
    float* projt = wts + o;
    {
        int total = OUTPUT * HIDDEN * 1;
        repack_w_kernel<<<(total + 255) / 256, 256, 0, stream>>>(
            projw, projt, OUTPUT, HIDDEN, 1);
        o += (size_t)total;
    }

    // ---- front end: scatter + FC -> bufX as (B,256,T) ----
    scatter_fc_kernel<<<dim3(TT, BB), 256, 0, stream>>>(x0, emb, fcw, fcb, bufX);

    const dim3 cgrid(TT / 32, BB);   // 32-column tiles

    // ---- 10 residual levels, 256 channels, dilation 2^i ----
    float* cur = bufX;
    float* oth = bufB;
    for (int i = 0; i < DEPTH; ++i) {
        int d = 1 << i;
        const float* b1 = (const float*)d_in[26 + 2 * i];
        const float* b2 = (const float*)d_in[27 + 2 * i];
        conv_wmma_kernel<HIDDEN, HIDDEN, 3, true, false, false>
            <<<cgrid, 256, 0, stream>>>(cur, w1t[i], b1, nullptr, bufA, d);
        conv_wmma_kernel<HIDDEN, HIDDEN, 3, true, true, false>
            <<<cgrid, 256, 0, stream>>>(bufA, w2t[i], b2, /*residual=*/cur, oth, d);
        float* tmp = cur; cur = oth; oth = tmp;
    }

    // ---- final level: 256 -> 320, dilation 2^10, projection residual ----
    {
        const float* b1 = (const float*)d_in[26 + 2 * DEPTH];
        const float* b2 = (const float*)d_in[27 + 2 * DEPTH];
        // res = 1x1 projection of cur (no gelu)
        conv_wmma_kernel<HIDDEN, OUTPUT, 1, false, false, false>
            <<<cgrid, 256, 0, stream>>>(cur, projt, projb, nullptr, bufR, 1);
        conv_wmma_kernel<HIDDEN, OUTPUT, 3, true, false, false>
            <<<cgrid, 256, 0, stream>>>(cur, w1t[DEPTH], b1, nullptr, bufA, 1 << DEPTH);
        // last conv: add residual, write transposed (B,T,320) straight to d_out
        conv_wmma_kernel<OUTPUT, OUTPUT, 3, true, true, true>
            <<<cgrid, 256, 0, stream>>>(bufA, w2t[DEPTH], b2, bufR, (float*)d_out,
                                        1 << DEPTH);
    }
}